// PromptParser_73469710565875
// MI455X (gfx1250) — compile-verified
//
#include <hip/hip_runtime.h>
#include <hip/hip_bf16.h>
#include <math.h>

typedef __attribute__((ext_vector_type(2))) float v2f;
typedef __attribute__((ext_vector_type(8))) float v8f;

#define NB 16   // batches
#define NT 16   // tokens (n)
#define NC 256  // embedding dim (c == x)

__device__ __forceinline__ float gelu_exact(float x) {
  return 0.5f * x * (1.0f + erff(x * 0.70710678118654752440f));
}

// att_m(c,x) = 3x3 cross-correlation of att with g (zero padded) + gb
__device__ __forceinline__ float conv3x3_at(const float* __restrict__ att, int c, int x,
                                            const float* __restrict__ g, float gb) {
  float acc = gb;
#pragma unroll
  for (int dc = -1; dc <= 1; ++dc) {
    int cc = c + dc;
    if (cc < 0 || cc >= NC) continue;
    const float* row = att + cc * NC;
#pragma unroll
    for (int dx = -1; dx <= 1; ++dx) {
      int xx = x + dx;
      if (xx < 0 || xx >= NC) continue;
      acc = fmaf(g[(dc + 1) * 3 + (dx + 1)], row[xx], acc);
    }
  }
  return acc;
}

// One workgroup per batch. 512 threads = 16 wave32s. LDS holds the whole
// per-batch working set (288KB of CDNA5's 320KB WGP LDS).
__global__ __launch_bounds__(512, 1)
void prompt_parser_fused(const float* __restrict__ img,
                         const float* __restrict__ tmp,
                         const float* __restrict__ p1,
                         const float* __restrict__ p2,
                         const float* __restrict__ w1,   // [3][12]
                         const float* __restrict__ b1,   // [12]
                         const float* __restrict__ w2,   // [12]
                         const float* __restrict__ b2,   // [1]
                         const float* __restrict__ gw,   // [3][3]
                         const float* __restrict__ gb,   // [1]
                         float* __restrict__ out_img,
                         float* __restrict__ out_res) {
  extern __shared__ float smem[];
  float* att    = smem;                 // [NC][NC]  256KB
  float* s_etpp = att + NC * NC;        // [NT][NC]   16KB
  float* s_t    = s_etpp + NT * NC;     // [NT][NC]   16KB

  const int b    = blockIdx.x;
  const int tid  = threadIdx.x;
  const int lane = tid & 31;
  const int wave = tid >> 5;            // 0..15
  const size_t base = (size_t)b * NT * NC;

  // ---------- Phase 0: elementwise MLP -> etpp, and t = tmp + p1 + p2 ----------
  {
    // Uniform-address weight loads -> scalar regs
    float rw1[36], rb1[12], rw2[12];
#pragma unroll
    for (int j = 0; j < 36; ++j) rw1[j] = w1[j];
#pragma unroll
    for (int j = 0; j < 12; ++j) { rb1[j] = b1[j]; rw2[j] = w2[j]; }
    const float rb2 = b2[0];

    for (int i = tid; i < NT * NC; i += blockDim.x) {
      float a  = tmp[base + i];
      float q1 = p1[base + i];
      float q2 = p2[base + i];
      s_t[i] = a + q1 + q2;                  // t = tmp + (p1 + p2)
      float acc = rb2;
#pragma unroll
      for (int j = 0; j < 12; ++j) {
        float h = fmaf(a, rw1[j], fmaf(q1, rw1[12 + j], fmaf(q2, rw1[24 + j], rb1[j])));
        acc = fmaf(gelu_exact(h), rw2[j], acc);
      }
      s_etpp[i] = acc;                       // layout [n][c]
      out_img[base + i] = img[base + i];     // passthrough output #1
    }
  }
  __syncthreads();

  // ---------- Phase 1: att[c][x] = sum_n etpp[n][c] * img[n][x] ----------
  // WMMA F32 16x16x4, K = NT = 16 -> 4 chained wmmas per 16x16 tile.
  // wave w owns c-tile w; loops over 16 x-tiles.
  {
    const int cTile = wave;
    const int m   = lane & 15;
    const int kb  = (lane >> 4) * 2;   // A/B fragment k-subindex (0 or 2)

    v2f afrag[4];
#pragma unroll
    for (int s = 0; s < 4; ++s) {
      int k0 = s * 4 + kb;             // token index n
      afrag[s].x = s_etpp[k0 * NC + cTile * 16 + m];
      afrag[s].y = s_etpp[(k0 + 1) * NC + cTile * 16 + m];
    }

    for (int xTile = 0; xTile < 16; ++xTile) {
      const int xcol = xTile * 16 + (lane & 15);
      v8f acc = {};
#pragma unroll
      for (int s = 0; s < 4; ++s) {
        int k0 = s * 4 + kb;
        v2f bfrag;
        bfrag.x = img[base + (size_t)k0 * NC + xcol];
        bfrag.y = img[base + (size_t)(k0 + 1) * NC + xcol];
        acc = __builtin_amdgcn_wmma_f32_16x16x4_f32(
            false, afrag[s], false, bfrag, (short)0, acc, false, false);
      }
      const int rhi = (lane >> 4) * 8;  // D layout: row m = r + 8*(lane>=16)
#pragma unroll
      for (int r = 0; r < 8; ++r)
        att[(cTile * 16 + r + rhi) * NC + xcol] = acc[r];
    }
  }
  __syncthreads();

  // ---------- Phase 2+3 fused: conv -> Mp/Mm -> 4 GEMMs (256x16, K=256) ----------
  // accA = Mp*tp2' + Mm*tm2'   (used where img >= 0)
  // accB = Mm*tp2' + Mp*tm2'   (used where img <  0)
  {
    float g9[9];
#pragma unroll
    for (int j = 0; j < 9; ++j) g9[j] = gw[j];
    const float gbv = gb[0];

    const int cTile = wave;
    const int m   = lane & 15;
    const int kb  = (lane >> 4) * 2;
    const int crow = cTile * 16 + m;

    v8f accA = {}, accB = {};
    for (int s = 0; s < 64; ++s) {
      const int kx0 = s * 4 + kb;          // x index
      // A fragment: clamped conv values at (crow, kx0) and (crow, kx0+1)
      float a0 = conv3x3_at(att, crow, kx0,     g9, gbv);
      float a1 = conv3x3_at(att, crow, kx0 + 1, g9, gbv);
      v2f mp = { fmaxf(a0, 1.0f), fmaxf(a1, 1.0f) };
      v2f mm = { fminf(a0, 1.0f), fminf(a1, 1.0f) };
      // B fragment: column n = lane&15, rows k = x; tp2/tm2 from t
      float t0 = s_t[(lane & 15) * NC + kx0];
      float t1 = s_t[(lane & 15) * NC + kx0 + 1];
      v2f btp = { (t0 >= 0.0f) ? t0 * t0 : 0.0f, (t1 >= 0.0f) ? t1 * t1 : 0.0f };
      v2f btm = { (t0 <  0.0f) ? t0 * t0 : 0.0f, (t1 <  0.0f) ? t1 * t1 : 0.0f };

      accA = __builtin_amdgcn_wmma_f32_16x16x4_f32(false, mp, false, btp, (short)0, accA, false, false);
      accA = __builtin_amdgcn_wmma_f32_16x16x4_f32(false, mm, false, btm, (short)0, accA, false, false);
      accB = __builtin_amdgcn_wmma_f32_16x16x4_f32(false, mm, false, btp, (short)0, accB, false, false);
      accB = __builtin_amdgcn_wmma_f32_16x16x4_f32(false, mp, false, btm, (short)0, accB, false, false);
    }

    // ---------- Epilogue: res[n][c] = img[n][c] * (img>=0 ? accA : accB) ----------
    const int ncol = lane & 15;            // token n
    const int rhi  = (lane >> 4) * 8;
#pragma unroll
    for (int r = 0; r < 8; ++r) {
      const int c = cTile * 16 + r + rhi;
      const float iv = img[base + (size_t)ncol * NC + c];
      const float sel = (iv >= 0.0f) ? accA[r] : accB[r];
      out_res[base + (size_t)ncol * NC + c] = iv * sel;
    }
  }
}

extern "C" void kernel_launch(void* const* d_in, const int* in_sizes, int n_in,
                              void* d_out, int out_size, void* d_ws, size_t ws_size,
                              hipStream_t stream) {
  const float* img = (const float*)d_in[0];
  const float* tmp = (const float*)d_in[1];
  const float* p1  = (const float*)d_in[2];
  const float* p2  = (const float*)d_in[3];
  const float* w1  = (const float*)d_in[4];
  const float* b1  = (const float*)d_in[5];
  const float* w2  = (const float*)d_in[6];
  const float* b2  = (const float*)d_in[7];
  const float* gw  = (const float*)d_in[8];
  const float* gb  = (const float*)d_in[9];

  float* out_img = (float*)d_out;
  float* out_res = (float*)d_out + NB * NT * NC;

  const size_t shmem = (size_t)(NC * NC + 2 * NT * NC) * sizeof(float); // 288KB
  (void)hipFuncSetAttribute((const void*)prompt_parser_fused,
                            hipFuncAttributeMaxDynamicSharedMemorySize, (int)shmem);

  hipLaunchKernelGGL(prompt_parser_fused, dim3(NB), dim3(512), shmem, stream,
                     img, tmp, p1, p2, w1, b1, w2, b2, gw, gb, out_img, out_res);
}